// MultiQueryAttention_42614665511260
// MI455X (gfx1250) — compile-verified
//
#include <hip/hip_runtime.h>
#include <hip/hip_bf16.h>

// ---------------------------------------------------------------------------
// MQA forward on gfx1250 via V_WMMA_F32_16X16X4_F32 (exact fp32 matrix path).
// B=2, N=2048, D=DK=DV=512, H=8.
// d_out = [ At : B*H*N*N floats ][ out : B*H*N*D floats ]
// The 'out' region is reused as staging for Q (identical element count),
// then overwritten by the final At@V GEMM.  d_ws holds K and V (16.8 MB).
// Wave tiles deepened to 32x64 (proj/av) and 16x128 (scores) so each
// A/B fragment feeds 8 WMMAs -> load:wmma issue ratio ~1:1.
// ---------------------------------------------------------------------------

typedef float v2f __attribute__((ext_vector_type(2)));
typedef float v8f __attribute__((ext_vector_type(8)));

#define BATCH 2
#define SEQ   2048
#define DIM   512
#define HEADS 8
#define SPITCH 2052   // 2048 + 4 pad: kills LDS bank conflicts on transposed pass

__device__ __forceinline__ v8f wmma4(v2f a, v2f b, v8f c) {
  // D = A(16x4,f32) * B(4x16,f32) + C(16x16,f32)
  return __builtin_amdgcn_wmma_f32_16x16x4_f32(
      /*neg_a=*/false, a, /*neg_b=*/false, b,
      /*c_mod=*/(short)0, c, /*reuse_a=*/false, /*reuse_b=*/false);
}

// ---------------------------------------------------------------------------
// C[m, n] = sum_k A[m,k] * W[n,k] + bias[n]      (A:[4096,512], W:[512,512])
// block = 256 thr (8 waves); wave tile 32x64; block tile 256x64.
// grid = (4096/256, 512/64)
// ---------------------------------------------------------------------------
__global__ void proj_nt_kernel(const float* __restrict__ A,
                               const float* __restrict__ W,
                               const float* __restrict__ bias,
                               float* __restrict__ C) {
  const int tid  = threadIdx.x;
  const int wave = tid >> 5;
  const int lane = tid & 31;
  const int c    = lane & 15;
  const int hi   = lane >> 4;
  const int rowBase = blockIdx.x * 256 + wave * 32;
  const int colBase = blockIdx.y * 64;

  v8f acc[2][4] = {};
  const float* ap0 = A + (size_t)(rowBase + c) * DIM + 2 * hi;
  const float* ap1 = ap0 + 16 * DIM;
  const float* bp[4];
#pragma unroll
  for (int t = 0; t < 4; ++t)
    bp[t] = W + (size_t)(colBase + 16 * t + c) * DIM + 2 * hi;

#pragma unroll 2
  for (int k0 = 0; k0 < DIM; k0 += 4) {
    v2f a0 = *(const v2f*)(ap0 + k0);
    v2f a1 = *(const v2f*)(ap1 + k0);
#pragma unroll
    for (int t = 0; t < 4; ++t) {
      v2f bb = *(const v2f*)(bp[t] + k0);
      acc[0][t] = wmma4(a0, bb, acc[0][t]);
      acc[1][t] = wmma4(a1, bb, acc[1][t]);
    }
  }

  float bv[4];
#pragma unroll
  for (int t = 0; t < 4; ++t) bv[t] = bias[colBase + 16 * t + c];
#pragma unroll
  for (int i = 0; i < 2; ++i) {
#pragma unroll
    for (int r = 0; r < 8; ++r) {
      const int row = rowBase + 16 * i + 8 * hi + r;
      float* cp = C + (size_t)row * DIM + colBase + c;
#pragma unroll
      for (int t = 0; t < 4; ++t) cp[16 * t] = acc[i][t][r] + bv[t];
    }
  }
}

// ---------------------------------------------------------------------------
// Q[b,h,n,k] = sum_d x[b,n,d] * Wq[h,k,d] + bq[h,k]
// grid = (4096/256, 512/64, H); wave tile 32x64.
// ---------------------------------------------------------------------------
__global__ void proj_q_kernel(const float* __restrict__ x,
                              const float* __restrict__ Wq,
                              const float* __restrict__ bq,
                              float* __restrict__ Q) {
  const int tid  = threadIdx.x;
  const int wave = tid >> 5;
  const int lane = tid & 31;
  const int c    = lane & 15;
  const int hi   = lane >> 4;
  const int h    = blockIdx.z;
  const int rowBase = blockIdx.x * 256 + wave * 32;
  const int colBase = blockIdx.y * 64;

  const float* W = Wq + (size_t)h * DIM * DIM;

  v8f acc[2][4] = {};
  const float* ap0 = x + (size_t)(rowBase + c) * DIM + 2 * hi;
  const float* ap1 = ap0 + 16 * DIM;
  const float* bp[4];
#pragma unroll
  for (int t = 0; t < 4; ++t)
    bp[t] = W + (size_t)(colBase + 16 * t + c) * DIM + 2 * hi;

#pragma unroll 2
  for (int k0 = 0; k0 < DIM; k0 += 4) {
    v2f a0 = *(const v2f*)(ap0 + k0);
    v2f a1 = *(const v2f*)(ap1 + k0);
#pragma unroll
    for (int t = 0; t < 4; ++t) {
      v2f bb = *(const v2f*)(bp[t] + k0);
      acc[0][t] = wmma4(a0, bb, acc[0][t]);
      acc[1][t] = wmma4(a1, bb, acc[1][t]);
    }
  }

  float bvv[4];
#pragma unroll
  for (int t = 0; t < 4; ++t) bvv[t] = bq[(size_t)h * DIM + colBase + 16 * t + c];
#pragma unroll
  for (int i = 0; i < 2; ++i) {
#pragma unroll
    for (int r = 0; r < 8; ++r) {
      const int m = rowBase + 16 * i + 8 * hi + r;   // m = b*SEQ + n
      const int b = m >> 11;
      const int n = m & (SEQ - 1);
      float* cp = Q + (((size_t)(b * HEADS + h) * SEQ) + n) * DIM + colBase + c;
#pragma unroll
      for (int t = 0; t < 4; ++t) cp[16 * t] = acc[i][t][r] + bvv[t];
    }
  }
}

// ---------------------------------------------------------------------------
// For a 16-query strip of head (b,h): S = Q_strip @ K^T / sqrt(DK),
// row softmax, transposed store At[m, n] = A[n, m].
// grid = (SEQ/16, B*H), block = 256 (8 waves). Each wave: 2 chunks of 128
// key columns (8 accumulators). LDS strip 16 x SPITCH f32 (~128 KB).
// ---------------------------------------------------------------------------
__global__ void scores_softmax_kernel(const float* __restrict__ Q,   // [B,H,N,DK]
                                      const float* __restrict__ Km,  // [B*N, DK]
                                      float* __restrict__ At) {      // [B*H,N,N]
  __shared__ float S[16 * SPITCH];

  const int tid  = threadIdx.x;
  const int wave = tid >> 5;
  const int lane = tid & 31;
  const int c    = lane & 15;
  const int hi   = lane >> 4;
  const int n0   = blockIdx.x * 16;
  const int bh   = blockIdx.y;
  const int b    = bh >> 3;

  const float* Qs = Q + ((size_t)bh * SEQ + n0) * DIM;   // [16, DIM] strip
  const float* Kb = Km + (size_t)b * SEQ * DIM;
  const float scale = 0.044194173824159216f;             // 1/sqrt(512)

  // ---- strip GEMM: each wave covers 2 chunks of 128 key columns ----
  for (int i = 0; i < 2; ++i) {
    const int m0 = wave * 128 + i * 1024;
    v8f acc[8] = {};
    const float* ap = Qs + (size_t)c * DIM + 2 * hi;
    const float* bp[8];
#pragma unroll
    for (int t = 0; t < 8; ++t)
      bp[t] = Kb + (size_t)(m0 + 16 * t + c) * DIM + 2 * hi;

#pragma unroll 2
    for (int k0 = 0; k0 < DIM; k0 += 4) {
      v2f a = *(const v2f*)(ap + k0);
#pragma unroll
      for (int t = 0; t < 8; ++t)
        acc[t] = wmma4(a, *(const v2f*)(bp[t] + k0), acc[t]);
    }
#pragma unroll
    for (int r = 0; r < 8; ++r) {
      float* sp = &S[(8 * hi + r) * SPITCH + m0 + c];
#pragma unroll
      for (int t = 0; t < 8; ++t) sp[16 * t] = acc[t][r] * scale;
    }
  }
  __syncthreads();

  // ---- softmax over keys: wave w owns rows 2w, 2w+1 ----
  for (int rr = 0; rr < 2; ++rr) {
    float* sp = &S[(wave * 2 + rr) * SPITCH];
    float mx = -3.402823466e38f;
    for (int m = lane; m < SEQ; m += 32) mx = fmaxf(mx, sp[m]);
#pragma unroll
    for (int o = 16; o > 0; o >>= 1) mx = fmaxf(mx, __shfl_xor(mx, o, 32));
    float sum = 0.0f;
    for (int m = lane; m < SEQ; m += 32) {
      float e = __expf(sp[m] - mx);
      sp[m] = e;
      sum += e;
    }
#pragma unroll
    for (int o = 16; o > 0; o >>= 1) sum += __shfl_xor(sum, o, 32);
    const float inv = 1.0f / sum;
    for (int m = lane; m < SEQ; m += 32) sp[m] *= inv;
  }
  __syncthreads();

  // ---- transposed store: At[bh, m, n0+cc] = S[cc][m] ----
  float* Ab = At + (size_t)bh * SEQ * SEQ + n0;
  const int cc = tid & 15;
  const int mb = tid >> 4;
  for (int m = mb; m < SEQ; m += 16) {
    Ab[(size_t)m * SEQ + cc] = S[cc * SPITCH + m];
  }
}

// ---------------------------------------------------------------------------
// out[bh, m, v] = sum_n At[bh, m, n] * V[b, n, v]    (NN GEMM, K-dim = SEQ)
// grid = (SEQ/256, DIM/64, B*H); wave tile 32x64.
// ---------------------------------------------------------------------------
__global__ void av_kernel(const float* __restrict__ At,  // [B*H, N, N]
                          const float* __restrict__ V,   // [B*N, DV]
                          float* __restrict__ Out) {     // [B*H, N, DV]
  const int tid  = threadIdx.x;
  const int wave = tid >> 5;
  const int lane = tid & 31;
  const int c    = lane & 15;
  const int hi   = lane >> 4;
  const int bh   = blockIdx.z;
  const int b    = bh >> 3;
  const int rowBase = blockIdx.x * 256 + wave * 32;
  const int colBase = blockIdx.y * 64;

  const float* Ap = At + (size_t)bh * SEQ * SEQ;
  const float* Vb = V + (size_t)b * SEQ * DIM;

  v8f acc[2][4] = {};
  const float* ap0 = Ap + (size_t)(rowBase + c) * SEQ + 2 * hi;
  const float* ap1 = ap0 + 16 * SEQ;

#pragma unroll 2
  for (int k0 = 0; k0 < SEQ; k0 += 4) {
    v2f a0 = *(const v2f*)(ap0 + k0);
    v2f a1 = *(const v2f*)(ap1 + k0);
    const float* v0 = Vb + (size_t)(k0 + 2 * hi) * DIM + colBase + c;
    const float* v1 = v0 + DIM;
#pragma unroll
    for (int t = 0; t < 4; ++t) {
      v2f bb = { v0[16 * t], v1[16 * t] };
      acc[0][t] = wmma4(a0, bb, acc[0][t]);
      acc[1][t] = wmma4(a1, bb, acc[1][t]);
    }
  }

#pragma unroll
  for (int i = 0; i < 2; ++i) {
#pragma unroll
    for (int r = 0; r < 8; ++r) {
      const size_t row = rowBase + 16 * i + 8 * hi + r;
      float* op = Out + ((size_t)bh * SEQ + row) * DIM + colBase + c;
#pragma unroll
      for (int t = 0; t < 4; ++t) op[16 * t] = acc[i][t][r];
    }
  }
}

// ---------------------------------------------------------------------------
extern "C" void kernel_launch(void* const* d_in, const int* in_sizes, int n_in,
                              void* d_out, int out_size, void* d_ws, size_t ws_size,
                              hipStream_t stream) {
  const float* x  = (const float*)d_in[0];
  const float* Wq = (const float*)d_in[1];
  const float* bq = (const float*)d_in[2];
  const float* Wk = (const float*)d_in[3];
  const float* bk = (const float*)d_in[4];
  const float* Wv = (const float*)d_in[5];
  const float* bv = (const float*)d_in[6];

  float* At   = (float*)d_out;
  float* OutQ = At + (size_t)BATCH * HEADS * SEQ * SEQ;  // out region; Q staging
  float* Kws  = (float*)d_ws;                            // [B*N, DK]
  float* Vws  = Kws + (size_t)BATCH * SEQ * DIM;         // [B*N, DV]

  const dim3 blk(256);
  proj_nt_kernel<<<dim3(BATCH * SEQ / 256, DIM / 64), blk, 0, stream>>>(x, Wk, bk, Kws);
  proj_nt_kernel<<<dim3(BATCH * SEQ / 256, DIM / 64), blk, 0, stream>>>(x, Wv, bv, Vws);
  proj_q_kernel<<<dim3(BATCH * SEQ / 256, DIM / 64, HEADS), blk, 0, stream>>>(x, Wq, bq, OutQ);
  scores_softmax_kernel<<<dim3(SEQ / 16, BATCH * HEADS), blk, 0, stream>>>(OutQ, Kws, At);
  av_kernel<<<dim3(SEQ / 256, DIM / 64, BATCH * HEADS), blk, 0, stream>>>(At, Vws, OutQ);
}